// SetCriterion_60232621359696
// MI455X (gfx1250) — compile-verified
//
#include <hip/hip_runtime.h>

typedef __attribute__((ext_vector_type(2))) float v2f;
typedef __attribute__((ext_vector_type(8))) float v8f;

#define MAXP 8732
#define TSTR 84   // padded tile row stride (floats); C must be <= 84

__device__ inline unsigned long long shfl_xor_u64(unsigned long long v, int m) {
  unsigned lo = (unsigned)v, hi = (unsigned)(v >> 32);
  lo = (unsigned)__shfl_xor((int)lo, m, 32);
  hi = (unsigned)__shfl_xor((int)hi, m, 32);
  return ((unsigned long long)hi << 32) | lo;
}

// ---------------- K1: matching + smooth-L1 box loss ----------------
__global__ __launch_bounds__(256) void k_match(
    const float* __restrict__ priors,     // P x 4 (cx,cy,w,h)
    const float* __restrict__ box_reg,    // B x P x 4
    const float* __restrict__ gt_boxes,   // B x G x 4 (xyxy)
    const int*   __restrict__ gt_labels,  // B x G
    int*   __restrict__ labels,           // B x P
    float* __restrict__ box_partial,      // B
    int*   __restrict__ pos_cnt,          // B
    int P, int G)
{
  const int b    = blockIdx.x;
  const int tid  = threadIdx.x;
  const int lane = tid & 31, wave = tid >> 5;

  __shared__ float         mval[MAXP];
  __shared__ unsigned char midx[MAXP];
  __shared__ float gtb[16 * 4];
  __shared__ int   gtl[16];
  __shared__ float gta[16];
  __shared__ unsigned long long redk[16][8];
  __shared__ float redf[256];
  __shared__ int   redi[256];

  if (tid < G * 4) gtb[tid] = gt_boxes[(size_t)b * G * 4 + tid];
  if (tid < G)     gtl[tid] = gt_labels[(size_t)b * G + tid];
  __syncthreads();
  if (tid < G) gta[tid] = (gtb[tid*4+2] - gtb[tid*4+0]) * (gtb[tid*4+3] - gtb[tid*4+1]);
  __syncthreads();

  unsigned long long bestk[16];
#pragma unroll
  for (int g = 0; g < 16; ++g) bestk[g] = 0ull;

  for (int p = tid; p < P; p += 256) {
    float cx = priors[p*4+0], cy = priors[p*4+1];
    float pw = priors[p*4+2], ph = priors[p*4+3];
    float px0 = cx - pw*0.5f, py0 = cy - ph*0.5f;
    float px1 = cx + pw*0.5f, py1 = cy + ph*0.5f;
    float pa  = pw * ph;
    float bv = -1.0f; int bg = 0;
#pragma unroll
    for (int g = 0; g < 16; ++g) {
      if (g >= G) break;
      float x0 = fmaxf(px0, gtb[g*4+0]);
      float y0 = fmaxf(py0, gtb[g*4+1]);
      float x1 = fminf(px1, gtb[g*4+2]);
      float y1 = fminf(py1, gtb[g*4+3]);
      float iw = fmaxf(x1 - x0, 0.f), ih = fmaxf(y1 - y0, 0.f);
      float inter = iw * ih;
      float iou = inter / (pa + gta[g] - inter);
      if (iou > bv) { bv = iou; bg = g; }  // strict > => first argmax
      // per-gt best prior key: (iou_bits << 32) | (~p) => max picks highest iou, then smallest p
      unsigned long long key = ((unsigned long long)__float_as_uint(iou) << 32)
                             | (unsigned)(0xFFFFFFFFu - (unsigned)p);
      bestk[g] = key > bestk[g] ? key : bestk[g];
    }
    mval[p] = bv;
    midx[p] = (unsigned char)bg;
  }

  // reduce per-gt keys: wave shuffle, then across 8 waves via LDS
#pragma unroll
  for (int g = 0; g < 16; ++g) {
    unsigned long long k = bestk[g];
    for (int off = 16; off > 0; off >>= 1) {
      unsigned long long o = shfl_xor_u64(k, off);
      k = o > k ? o : k;
    }
    if (lane == 0) redk[g][wave] = k;
  }
  __syncthreads();
  if (tid < G) {
    unsigned long long k = redk[tid][0];
    for (int w = 1; w < 8; ++w) { unsigned long long o = redk[tid][w]; k = o > k ? o : k; }
    unsigned bp = 0xFFFFFFFFu - (unsigned)(k & 0xFFFFFFFFu);
    if (bp < (unsigned)P) { midx[bp] = (unsigned char)tid; mval[bp] = 2.0f; }
  }
  __syncthreads();

  // phase 2: labels, pos count, encode + smooth-L1
  float boxsum = 0.f; int npos = 0;
  for (int p = tid; p < P; p += 256) {
    float v = mval[p]; int g = midx[p];
    int lab = (v < 0.5f) ? 0 : gtl[g];
    labels[(size_t)b * P + p] = lab;
    if (lab > 0) {
      npos++;
      float bx0 = gtb[g*4+0], by0 = gtb[g*4+1], bx1 = gtb[g*4+2], by1 = gtb[g*4+3];
      float cx = priors[p*4+0], cy = priors[p*4+1];
      float pw = priors[p*4+2], ph = priors[p*4+3];
      float t0 = ((bx0 + bx1) * 0.5f - cx) / (0.1f * pw);
      float t1 = ((by0 + by1) * 0.5f - cy) / (0.1f * ph);
      float t2 = logf((bx1 - bx0) / pw) / 0.2f;
      float t3 = logf((by1 - by0) / ph) / 0.2f;
      const float* r = box_reg + ((size_t)b * P + p) * 4;
      float t[4] = {t0, t1, t2, t3};
#pragma unroll
      for (int i = 0; i < 4; ++i) {
        float d = fabsf(r[i] - t[i]);
        boxsum += (d < 1.f) ? 0.5f * d * d : d - 0.5f;
      }
    }
  }
  redf[tid] = boxsum; redi[tid] = npos; __syncthreads();
  for (int s = 128; s > 0; s >>= 1) {
    if (tid < s) { redf[tid] += redf[tid + s]; redi[tid] += redi[tid + s]; }
    __syncthreads();
  }
  if (tid == 0) { box_partial[b] = redf[0]; pos_cnt[b] = redi[0]; }
}

// ---------------- K2: log-softmax via LDS staging + WMMA f32 16x16x4 row-sum ----------------
// One wave owns 16 rows. Cooperative coalesced load of the 16xC tile into LDS (stride 84 =>
// conflict-free reads for the WMMA feed pattern), single HBM pass, then:
//   pass A: row max from LDS; pass B: exp + WMMA accumulate (A-layout: lanes 0-15 feed
//   K=(4i,4i+1), lanes 16-31 feed K=(4i+2,4i+3); B = ones => accumulator columns = row sums).
__global__ __launch_bounds__(256) void k_softmax(
    const float* __restrict__ logits,  // R rows x C
    const int*   __restrict__ labels,  // R
    float* __restrict__ bg_loss,       // R
    float* __restrict__ ce,            // R
    long long R, int C)
{
  __shared__ float tile[8][16 * TSTR];   // 43 KB
  const int tid  = threadIdx.x;
  const int lane = tid & 31;
  const int w    = tid >> 5;
  const long long base = ((long long)blockIdx.x * 8 + w) * 16;
  const bool active = base < R;
  float* tl = tile[w];

  if (active) {
#pragma unroll 4
    for (int r = 0; r < 16; ++r) {
      long long rr = base + r; if (rr >= R) rr = R - 1;
      const float* src = logits + rr * (long long)C;
      for (int c = lane; c < C; c += 32) tl[r * TSTR + c] = src[c];
    }
  }
  __syncthreads();
  if (!active) return;   // after the only barrier; wave-uniform

  const int myrow = lane & 15;
  const int rbase = myrow * TSTR;
  const int coff  = (lane < 16) ? 0 : 2;
  const int kfull = C >> 2;          // 20 for C=81
  const int ctail = kfull << 2;      // 80

  // pass A: row max (no guards)
  float m = -INFINITY;
  for (int i = 0; i < kfull; ++i) {
    int c0 = i * 4 + coff;
    m = fmaxf(m, tl[rbase + c0]);
    m = fmaxf(m, tl[rbase + c0 + 1]);
  }
  for (int c = ctail; c < C; ++c) m = fmaxf(m, tl[rbase + c]);
  m = fmaxf(m, __shfl_xor(m, 16, 32));

  // pass B: exp + WMMA accumulate (no guards; tail column added after extraction)
  v8f acc = {0.f, 0.f, 0.f, 0.f, 0.f, 0.f, 0.f, 0.f};
  v2f bones; bones.x = 1.0f; bones.y = 1.0f;
  for (int i = 0; i < kfull; ++i) {
    int c0 = i * 4 + coff;
    v2f a;
    a.x = __expf(tl[rbase + c0]     - m);
    a.y = __expf(tl[rbase + c0 + 1] - m);
    acc = __builtin_amdgcn_wmma_f32_16x16x4_f32(false, a, false, bones,
                                                (short)0, acc, false, false);
  }

  // extract row sums: row r is in VGPR (r&7) on lane-group (r<8 ? 0-15 : 16-31)
  float t;
  switch (lane & 7) {
    case 0: t = acc[0]; break; case 1: t = acc[1]; break;
    case 2: t = acc[2]; break; case 3: t = acc[3]; break;
    case 4: t = acc[4]; break; case 5: t = acc[5]; break;
    case 6: t = acc[6]; break; default: t = acc[7]; break;
  }
  int srcl = (myrow < 8) ? myrow : (myrow + 8);
  float s = __shfl(t, srcl, 32);

  long long row = base + myrow;
  if (lane < 16 && row < R) {
    for (int c = ctail; c < C; ++c) s += __expf(tl[rbase + c] - m);  // tail column(s)
    float lse = m + logf(s);
    int lab = labels[row];
    bg_loss[row] = lse - tl[rbase];
    ce[row]      = lse - tl[rbase + lab];
  }
}

// ---------------- K3: hard-negative mining via radix-select + masked ce sum ----------------
__global__ __launch_bounds__(256) void k_mine(
    const float* __restrict__ bg_loss,
    const float* __restrict__ ce,
    const int*   __restrict__ labels,
    const int*   __restrict__ pos_cnt,
    float* __restrict__ cls_partial,
    int P)
{
  const int b = blockIdx.x; const int tid = threadIdx.x;
  __shared__ unsigned u_s[MAXP];
  __shared__ unsigned hist[256];
  __shared__ unsigned sh_prefix; __shared__ int sh_k;
  __shared__ float redf[256]; __shared__ int redi[256];

  const size_t base = (size_t)b * P;
  for (int p = tid; p < P; p += 256) {
    int lab = labels[base + p];
    unsigned u = 0u;                               // positives -> -inf -> minimum key
    if (lab <= 0) {
      unsigned bits = __float_as_uint(bg_loss[base + p]);
      u = (bits & 0x80000000u) ? ~bits : (bits | 0x80000000u);  // order-preserving map
      if (u == 0u) u = 1u;
    }
    u_s[p] = u;
  }
  int nn = pos_cnt[b] * 3;
  if (nn > P) nn = P;
  if (tid == 0) { sh_prefix = 0u; sh_k = nn; }
  __syncthreads();

  unsigned ut = 0u; int quota_eq = 0;
  if (nn > 0) {
    for (int d = 3; d >= 0; --d) {
      for (int i = tid; i < 256; i += 256) hist[i] = 0u;
      __syncthreads();
      unsigned pmask = (d == 3) ? 0u : (0xFFFFFFFFu << ((d + 1) * 8));
      unsigned pref  = sh_prefix;
      for (int p = tid; p < P; p += 256) {
        unsigned u = u_s[p];
        if ((u & pmask) == (pref & pmask))
          atomicAdd(&hist[(u >> (d * 8)) & 0xFFu], 1u);
      }
      __syncthreads();
      if (tid == 0) {
        int k = sh_k; int bin = 255;
        for (; bin > 0; --bin) {
          int h = (int)hist[bin];
          if (k <= h) break;
          k -= h;
        }
        sh_prefix |= ((unsigned)bin) << (d * 8);
        sh_k = k;
      }
      __syncthreads();
    }
    ut = sh_prefix;
    int lg = 0;
    for (int p = tid; p < P; p += 256) if (u_s[p] > ut) lg++;
    redi[tid] = lg; __syncthreads();
    for (int s = 128; s > 0; s >>= 1) { if (tid < s) redi[tid] += redi[tid + s]; __syncthreads(); }
    quota_eq = nn - redi[0];
    __syncthreads();
  }

  float local = 0.f;
  for (int p = tid; p < P; p += 256) {
    int lab = labels[base + p];
    if (lab > 0) { local += ce[base + p]; continue; }   // positives always in mask
    if (nn <= 0) continue;
    unsigned u = u_s[p];
    if (u > ut) {
      local += ce[base + p];
    } else if (u == ut) {
      int before = 0;                                   // stable tie-break by index
      for (int q = 0; q < p; ++q) if (u_s[q] == ut) before++;
      if (before < quota_eq) local += ce[base + p];
    }
  }
  redf[tid] = local; __syncthreads();
  for (int s = 128; s > 0; s >>= 1) { if (tid < s) redf[tid] += redf[tid + s]; __syncthreads(); }
  if (tid == 0) cls_partial[b] = redf[0];
}

// ---------------- K4: deterministic fixed-order final reduction ----------------
__global__ void k_final(const float* __restrict__ box_partial,
                        const float* __restrict__ cls_partial,
                        const int*   __restrict__ pos_cnt,
                        float* __restrict__ out, int B)
{
  if (threadIdx.x == 0 && blockIdx.x == 0) {
    float bs = 0.f, cs = 0.f; int np = 0;
    for (int i = 0; i < B; ++i) { bs += box_partial[i]; cs += cls_partial[i]; np += pos_cnt[i]; }
    float inv = 1.0f / (float)np;
    out[0] = bs * inv;
    out[1] = cs * inv;
  }
}

extern "C" void kernel_launch(void* const* d_in, const int* in_sizes, int n_in,
                              void* d_out, int out_size, void* d_ws, size_t ws_size,
                              hipStream_t stream) {
  const float* priors         = (const float*)d_in[0];
  const float* class_logits   = (const float*)d_in[1];
  const float* box_regression = (const float*)d_in[2];
  const float* gt_boxes       = (const float*)d_in[3];
  const int*   gt_labels      = (const int*)d_in[4];

  const int P = in_sizes[0] / 4;                                // 8732
  const int B = (int)((long long)in_sizes[2] / (4LL * P));      // 64
  const int G = in_sizes[4] / B;                                // 16
  const int C = (int)((long long)in_sizes[1] / ((long long)B * P)); // 81
  const long long R = (long long)B * P;

  char* ws = (char*)d_ws;
  int*   labels = (int*)ws;   ws += (size_t)R * sizeof(int);
  float* bg     = (float*)ws; ws += (size_t)R * sizeof(float);
  float* cearr  = (float*)ws; ws += (size_t)R * sizeof(float);
  float* boxp   = (float*)ws; ws += (size_t)B * sizeof(float);
  float* clsp   = (float*)ws; ws += (size_t)B * sizeof(float);
  int*   posc   = (int*)ws;   ws += (size_t)B * sizeof(int);

  k_match<<<B, 256, 0, stream>>>(priors, box_regression, gt_boxes, gt_labels,
                                 labels, boxp, posc, P, G);

  long long waves  = (R + 15) / 16;
  int       blocks = (int)((waves + 7) / 8);
  k_softmax<<<blocks, 256, 0, stream>>>(class_logits, labels, bg, cearr, R, C);

  k_mine<<<B, 256, 0, stream>>>(bg, cearr, labels, posc, clsp, P);

  k_final<<<1, 64, 0, stream>>>(boxp, clsp, posc, (float*)d_out, B);
}